// ViewSynthesisNetwork_1726576853206
// MI455X (gfx1250) — compile-verified
//
#include <hip/hip_runtime.h>

// Problem constants (from reference setup_inputs)
#define NDISP 65
#define Bn 2
#define Cn 3
#define Hn 384
#define Wn 1280
#define HWc ((size_t)Hn * Wn)

#define TILE_W 16
#define WAVES 4                   // waves per workgroup (wave32)
#define HALF_W 640                // columns handled per workgroup
#define TILES (HALF_W / TILE_W)   // 40 tiles per workgroup

#define VPITCH 1360               // padded view row pitch in floats (1280 + 80 zero pad)
#define DROWS 96                  // disp tile rows: 16 guard + 65 data + 15 guard
#define DOFF 16                   // LDS row index of disparity plane d=0
#define DSLAB (DROWS * TILE_W)    // floats per disp tile buffer (1536)

#define LDS_FLOATS (3 * VPITCH + WAVES * 2 * DSLAB)   // 16368 floats = 65472 B

typedef __attribute__((ext_vector_type(2))) float v2f;
typedef __attribute__((ext_vector_type(8))) float v8f;
typedef int v4i __attribute__((vector_size(4 * sizeof(int))));  // matches builtin param

#define AS1 __attribute__((address_space(1)))
#define AS3 __attribute__((address_space(3)))

#if defined(__has_builtin)
#  if __has_builtin(__builtin_amdgcn_global_load_async_to_lds_b128)
#    define ASYNC_PATH 1
#  endif
#endif
#ifndef ASYNC_PATH
#  define ASYNC_PATH 0
#endif

__device__ __forceinline__ void wait_async_le9() {
#if ASYNC_PATH
#  if __has_builtin(__builtin_amdgcn_s_wait_asynccnt)
    __builtin_amdgcn_s_wait_asynccnt(9);
#  else
    asm volatile("s_wait_asynccnt 0x9" ::: "memory");
#  endif
#endif
}
__device__ __forceinline__ void wait_async_0() {
#if ASYNC_PATH
#  if __has_builtin(__builtin_amdgcn_s_wait_asynccnt)
    __builtin_amdgcn_s_wait_asynccnt(0);
#  else
    asm volatile("s_wait_asynccnt 0x0" ::: "memory");
#  endif
#endif
}

__global__ __launch_bounds__(WAVES * 32)
void vsynth_wmma_kernel(const float* __restrict__ disp,
                        const float* __restrict__ view,
                        float* __restrict__ out)
{
    // LDS: [0, 3*VPITCH)  padded view rows c=0..2 (tail 80 floats zero)
    //      then WAVES * 2 double-buffered disp tile slabs (zero guard rows)
    __shared__ float lds[LDS_FLOATS];
    float* Lview = lds;
    float* Ldisp = lds + 3 * VPITCH;

    const int tid  = threadIdx.x;
    const int lane = tid & 31;
    const int wv   = tid >> 5;
    const int j    = lane & 15;   // WMMA column (B/D) and M row (A)
    const int half = lane >> 4;   // selects K pair {0,1} vs {2,3}
    const int r    = lane >> 2;   // staging: row within group of 8
    const int cb   = (lane & 3) * 4;  // staging: float column base

    const int bh = blockIdx.x;
    const int b  = bh / Hn;
    const int h  = bh % Hn;
    const int wstart = blockIdx.y * HALF_W;

    const float* dispBase = disp + (size_t)b * NDISP * HWc + (size_t)h * Wn;
    const float* viewBase = view + (size_t)b * Cn * HWc + (size_t)h * Wn;
    float*       outBase  = out  + (size_t)b * Cn * HWc + (size_t)h * Wn;

    // ---- zero all LDS (view pad tail + disp guard rows; data rows overwritten) ----
    {
        float4 z = make_float4(0.f, 0.f, 0.f, 0.f);
        for (int i = tid; i < LDS_FLOATS / 4; i += WAVES * 32)
            reinterpret_cast<float4*>(lds)[i] = z;
    }
    __syncthreads();

    // ---- stage the 3 view rows for this (b,h) into LDS (coalesced b128) ----
    for (int i = tid; i < (Cn * Wn) / 4; i += WAVES * 32) {
        int c = i / (Wn / 4);
        int q = i % (Wn / 4);
        float4 v = reinterpret_cast<const float4*>(viewBase + (size_t)c * HWc)[q];
        reinterpret_cast<float4*>(Lview + c * VPITCH)[q] = v;
    }
    __syncthreads();

    // A-operand row: channel j for j<3; lanes 3..15 harmlessly duplicate row 2
    // (D rows 3..15 are never stored and cannot affect rows 0..2).
    const float* Aview = Lview + (j < Cn ? j : 2) * VPITCH;

    // Stage one 65x16 disp tile into LDS buffer `buf` (rows DOFF..DOFF+64).
    // 9 b128 groups; group 8 clamps to row 64 (duplicate same-address same-data).
    auto stage = [&](int tile, int buf) {
        float* LDb = Ldisp + (wv * 2 + buf) * DSLAB;
        const int w0 = wstart + tile * TILE_W;
#if ASYNC_PATH
#pragma unroll
        for (int i = 0; i < 9; ++i) {
            int d = i * 8 + r;  if (d > 64) d = 64;
            const float* gp = dispBase + (size_t)d * HWc + w0 + cb;
            float*       lp = LDb + (DOFF + d) * TILE_W + cb;
            __builtin_amdgcn_global_load_async_to_lds_b128(
                (AS1 v4i*)gp, (AS3 v4i*)lp, /*offset=*/0, /*cpol=*/0);
        }
#else
        float4 tmp[9];
#pragma unroll
        for (int i = 0; i < 9; ++i) {
            int d = i * 8 + r;  if (d > 64) d = 64;
            tmp[i] = *reinterpret_cast<const float4*>(dispBase + (size_t)d * HWc + w0 + cb);
        }
#pragma unroll
        for (int i = 0; i < 9; ++i) {
            int d = i * 8 + r;  if (d > 64) d = 64;
            *reinterpret_cast<float4*>(LDb + (DOFF + d) * TILE_W + cb) = tmp[i];
        }
#endif
    };

    // ---- software-pipelined tile loop (double-buffered LDS) ----
    stage(wv, 0);
    int buf = 0;
    for (int tile = wv; tile < TILES; tile += WAVES) {
        const int nxt = tile + WAVES;
        if (nxt < TILES) {
            stage(nxt, buf ^ 1);       // issue async group for next tile
            wait_async_le9();          // <=9 outstanding => current buffer complete
        } else {
            wait_async_0();
        }
        if (tile + 2 * WAVES < TILES) {  // warm L2 two tiles ahead
            const float* p0 = dispBase + (size_t)lane        * HWc + (wstart + (tile + 2 * WAVES) * TILE_W);
            const float* p1 = dispBase + (size_t)(lane + 32) * HWc + (wstart + (tile + 2 * WAVES) * TILE_W);
            __builtin_prefetch(p0, 0, 1);
            __builtin_prefetch(p1, 0, 1);
        }

        const float* LDb = Ldisp + (wv * 2 + buf) * DSLAB;
        const int w0 = wstart + tile * TILE_W;

        // banded GEMM: out(3x16) += view(3x4) x M(4x16), 20 K-chunks of 4
        v8f acc0 = {};
        v8f acc1 = {};
#pragma unroll
        for (int t = 0; t < 20; ++t) {
            // A (16x4 f32): lane = M row; VGPR0/1 hold K = 2*half + {0,1}
            v2f a = *reinterpret_cast<const v2f*>(Aview + (w0 + 4 * t + 2 * half));
            // B (4x16 f32): lane = N col j; M[w', w0+j] = disp[w'-(w0+j), w0+j]
            int drow = DOFF + 4 * t + 2 * half - j;   // in [1, 95]; guards are zero
            v2f bm;
            bm.x = LDb[drow * TILE_W + j];
            bm.y = LDb[(drow + 1) * TILE_W + j];
            if (t & 1)
                acc1 = __builtin_amdgcn_wmma_f32_16x16x4_f32(
                    false, a, false, bm, (short)0, acc1, false, false);
            else
                acc0 = __builtin_amdgcn_wmma_f32_16x16x4_f32(
                    false, a, false, bm, (short)0, acc0, false, false);
        }
        v8f acc = acc0 + acc1;

        // D layout: VGPR c holds row M=c in lanes 0..15 (lanes 16..31 unused)
        if (lane < 16) {
            float* ob = outBase + w0 + lane;
            ob[0 * HWc] = acc[0];
            ob[1 * HWc] = acc[1];
            ob[2 * HWc] = acc[2];
        }
        buf ^= 1;
    }
}

extern "C" void kernel_launch(void* const* d_in, const int* in_sizes, int n_in,
                              void* d_out, int out_size, void* d_ws, size_t ws_size,
                              hipStream_t stream) {
    (void)in_sizes; (void)n_in; (void)out_size; (void)d_ws; (void)ws_size;
    const float* disp = (const float*)d_in[0];   // [2,65,384,1280] f32
    const float* view = (const float*)d_in[1];   // [2,3,384,1280] f32
    float* out = (float*)d_out;                  // [2,3,384,1280] f32

    dim3 grid(Bn * Hn, Wn / HALF_W);   // (768, 2)
    dim3 block(WAVES * 32);            // 128 threads = 4 wave32
    vsynth_wmma_kernel<<<grid, block, 0, stream>>>(disp, view, out);
}